// LJ_4647154614865
// MI455X (gfx1250) — compile-verified
//
#include <hip/hip_runtime.h>

#define N_ATOMS 6144
#define CELLF   66.0f
#define SIGMAF  3.405f
#define EPSF    0.0103f
#define CUTF    9.0f
#define TILE    16
#define TILES   (N_ATOMS / TILE)   // 384
#define WAVES_PER_BLOCK 8

typedef __attribute__((ext_vector_type(2))) float v2f;
typedef __attribute__((ext_vector_type(8))) float v8f;

// One wave32 computes one 16x16 tile of the pair-energy matrix.
// r^2 = |qi|^2 + |qj|^2 - 2*dot(qi,qj) + per-dim minimum-image correction.
// The Gram term dot(qi,qj) for the whole tile is a single
// V_WMMA_F32_16X16X4_F32 (K = {x,y,z,0}).
__global__ __launch_bounds__(256) void lj_tile_kernel(
    const float* __restrict__ q, float* __restrict__ out)
{
    const int lane = threadIdx.x & 31;
    const int wave = threadIdx.x >> 5;
    const int tile = blockIdx.x * WAVES_PER_BLOCK + wave;
    const int ti = tile / TILES;   // row block
    const int tj = tile % TILES;   // col block

    const int half = lane >> 4;    // 0: lanes 0-15, 1: lanes 16-31
    const int l15  = lane & 15;

    // Column this lane owns (same for all 8 C VGPRs).
    const int j = tj * TILE + l15;
    const float qjx = q[3 * j + 0];
    const float qjy = q[3 * j + 1];
    const float qjz = q[3 * j + 2];

    if (ti > tj) {
        // Strictly-lower tile: all zeros, no compute. Wave-uniform branch,
        // so EXEC is all-1s on the WMMA path below.
        #pragma unroll
        for (int v = 0; v < 8; ++v) {
            const int i = ti * TILE + half * 8 + v;
            __builtin_nontemporal_store(0.0f, &out[(size_t)i * N_ATOMS + j]);
        }
        return;
    }

    // ---- Build A (16x4) / B (4x16) fragments for V_WMMA_F32_16X16X4_F32 ----
    // A layout (32-bit, 16x4): lanes 0-15 -> row M=lane, VGPR0=K0, VGPR1=K1;
    //                          lanes 16-31 -> row M=lane-16, VGPR0=K2, VGPR1=K3.
    const int arow = ti * TILE + l15;
    float a0, a1, b0, b1;
    if (half == 0) {
        a0 = q[3 * arow + 0];  a1 = q[3 * arow + 1];   // K0=x, K1=y
        b0 = qjx;              b1 = qjy;
    } else {
        a0 = q[3 * arow + 2];  a1 = 0.0f;              // K2=z, K3=pad(0)
        b0 = qjz;              b1 = 0.0f;
    }
    v2f A = {a0, a1};
    v2f B = {b0, b1};
    v8f C = {};
    // D = A x B + 0 : C[v] = dot(q_i, q_j), i = ti*16 + v + 8*half, j as above
    C = __builtin_amdgcn_wmma_f32_16x16x4_f32(
        /*neg_a=*/false, A, /*neg_b=*/false, B,
        /*c_mod=*/(short)0, C, /*reuse_a=*/false, /*reuse_b=*/false);

    const float sj       = qjx * qjx + qjy * qjy + qjz * qjz;
    const float halfcell = 0.5f * CELLF;
    const float cell2    = CELLF * CELLF;
    const float cut2     = CUTF * CUTF;
    const float sig2     = SIGMAF * SIGMAF;
    const float fourEps  = 4.0f * EPSF;

    float res[8];
    #pragma unroll
    for (int v = 0; v < 8; ++v) {
        const int i = ti * TILE + half * 8 + v;
        const float qix = q[3 * i + 0];
        const float qiy = q[3 * i + 1];
        const float qiz = q[3 * i + 2];
        const float si  = qix * qix + qiy * qiy + qiz * qiz;

        // Unwrapped squared distance from the Gram term.
        float r2 = si + sj - 2.0f * C[v];

        // Minimum-image correction per dimension: coords in [0,CELL) so
        // |d| < CELL; wrap fires iff |d| > CELL/2, adding (|d|-CELL)^2 - d^2
        // = CELL^2 - 2*CELL*|d|.
        const float dx = fabsf(qix - qjx);
        const float dy = fabsf(qiy - qjy);
        const float dz = fabsf(qiz - qjz);
        r2 += (dx > halfcell) ? fmaf(-2.0f * CELLF, dx, cell2) : 0.0f;
        r2 += (dy > halfcell) ? fmaf(-2.0f * CELLF, dy, cell2) : 0.0f;
        r2 += (dz > halfcell) ? fmaf(-2.0f * CELLF, dz, cell2) : 0.0f;

        // LJ energy: 4*eps*((s2/r2)^6 - (s2/r2)^3) with s2 = sigma^2.
        const float inv = sig2 / r2;            // diag r2==0 -> NaN, masked below
        const float sr6 = inv * inv * inv;
        const float e   = fourEps * fmaf(sr6, sr6, -sr6);

        const bool ok = (r2 < cut2) && (i < j);
        res[v] = ok ? e : 0.0f;
    }

    #pragma unroll
    for (int v = 0; v < 8; ++v) {
        const int i = ti * TILE + half * 8 + v;
        __builtin_nontemporal_store(res[v], &out[(size_t)i * N_ATOMS + j]);
    }
}

extern "C" void kernel_launch(void* const* d_in, const int* in_sizes, int n_in,
                              void* d_out, int out_size, void* d_ws, size_t ws_size,
                              hipStream_t stream) {
    const float* q = (const float*)d_in[0];
    float* out = (float*)d_out;

    const int total_tiles = TILES * TILES;              // 147456
    const int blocks = total_tiles / WAVES_PER_BLOCK;   // 18432
    lj_tile_kernel<<<blocks, 256, 0, stream>>>(q, out);
}